// NeuralSDE_57526791963286
// MI455X (gfx1250) — compile-verified
//
#include <hip/hip_runtime.h>
#include <cstdint>

// Problem constants (match reference)
#define T_STEPS 100
#define N_STEPS 99            // T-1 integration steps
#define BD      (2048 * 512)  // B*D elements
#define BLOCK   256
#define VEC     4
#define NB      4             // LDS ring buffers
#define DEPTH   3             // async prefetch depth (uniform s_wait_asynccnt 0x2)

static_assert(BD * 4 == (1 << 22), "noise step stride must be 1<<22 bytes");
static_assert(BLOCK * 16 == (1 << 12), "LDS buffer stride must be 1<<12 bytes");

// Native clang vector type (usable with __builtin_nontemporal_store)
typedef float v4f __attribute__((ext_vector_type(4)));

// ---------------------------------------------------------------------------
// Setup kernel: precompute per-step uniform constants {sin t, cos t, dt, sqrt(dt)}
// (one-shot, 99 threads; precise libm is fine here).
// ---------------------------------------------------------------------------
__global__ void sde_setup(const float* __restrict__ ts, v4f* __restrict__ cst) {
    int i = blockIdx.x * blockDim.x + threadIdx.x;
    if (i < N_STEPS) {
        float t  = ts[i];
        float dt = ts[i + 1] - t;
        v4f c;
        c.x = sinf(t);
        c.y = cosf(t);
        c.z = dt;
        c.w = sqrtf(dt);
        cst[i] = c;
    }
}

__device__ __forceinline__ float sde_step(float yi, float zi, float sin_t,
                                          float cos_t, float dt, float sqdt,
                                          float th) {
    float ex = __expf(-yi);
    float x  = cos_t * ex;
    float sg = 1.0f / (1.0f + __expf(-x));     // sigmoid
    float drift = sin_t + th * yi;
    return yi + drift * dt + (0.3f * sg) * sqdt * zi;
}

// ---------------------------------------------------------------------------
// Main kernel: one thread per 4 consecutive elements, loop over 99 steps.
// Noise streamed via async global->LDS, 4-buffer ring, 3-deep prefetch.
// ---------------------------------------------------------------------------
__global__ __launch_bounds__(BLOCK) void sde_main(
    const float* __restrict__ y0,
    const float* __restrict__ noise,
    const float* __restrict__ theta_p,
    const v4f*   __restrict__ cst,
    float*       __restrict__ out)
{
    __shared__ v4f buf[NB][BLOCK];   // 16 KB ring buffer per workgroup

    const int      tid = threadIdx.x;
    const unsigned gid = blockIdx.x * BLOCK + tid;
    const unsigned e0  = gid * VEC;

    // Workgroup-relative LDS byte address of this lane's slot in buffer 0
    const unsigned ldsb = (unsigned)(uintptr_t)&buf[0][tid];
    // Per-lane byte offset into noise at step 0 (GVS: saddr64 + vaddr32)
    const unsigned gb0  = e0 * 4u;

    // Prologue: fill the pipeline with steps 0..DEPTH-1 (ASYNCcnt = 3)
#pragma unroll
    for (unsigned k = 0; k < DEPTH; ++k) {
        unsigned lds = ldsb + (k << 12);
        unsigned gb  = gb0 + (k << 22);
        asm volatile("global_load_async_to_lds_b128 %0, %1, %2"
                     :: "v"(lds), "v"(gb), "s"(noise) : "memory");
    }

    const float th = theta_p[0];

    v4f y = *(const v4f*)(y0 + e0);
    // Trajectory row 0 = y0 (streaming store, don't pollute L2)
    __builtin_nontemporal_store(y, (v4f*)(out + e0));

    for (int s = 0; s < N_STEPS; ++s) {
        v4f c = cst[s];                 // uniform: {sin t, cos t, dt, sqrt(dt)}

        // In-order completion: <=2 outstanding => buffer for step s has landed
        asm volatile("s_wait_asynccnt 0x2" ::: "memory");
        v4f z = buf[s & (NB - 1)][tid];
        // DS read must have sampled LDS before the async engine may
        // overwrite a ring slot (WAR guard), then keep pipeline at depth 3.
        asm volatile("s_wait_dscnt 0x0" ::: "memory");

        unsigned ps  = (unsigned)(s + DEPTH);
        if (ps > N_STEPS - 1) ps = N_STEPS - 1;          // clamp tail
        unsigned lds = ldsb + (((unsigned)(s + DEPTH) & (NB - 1)) << 12);
        unsigned gb  = gb0 + (ps << 22);
        asm volatile("global_load_async_to_lds_b128 %0, %1, %2"
                     :: "v"(lds), "v"(gb), "s"(noise) : "memory");

        y.x = sde_step(y.x, z.x, c.x, c.y, c.z, c.w, th);
        y.y = sde_step(y.y, z.y, c.x, c.y, c.z, c.w, th);
        y.z = sde_step(y.z, z.z, c.x, c.y, c.z, c.w, th);
        y.w = sde_step(y.w, z.w, c.x, c.y, c.z, c.w, th);

        __builtin_nontemporal_store(
            y, (v4f*)(out + (size_t)(s + 1) * BD + e0));
    }
}

// ---------------------------------------------------------------------------
extern "C" void kernel_launch(void* const* d_in, const int* in_sizes, int n_in,
                              void* d_out, int out_size, void* d_ws, size_t ws_size,
                              hipStream_t stream) {
    const float* y0    = (const float*)d_in[0];
    const float* ts    = (const float*)d_in[1];
    const float* noise = (const float*)d_in[2];
    const float* theta = (const float*)d_in[3];
    float*       out   = (float*)d_out;
    v4f*         cst   = (v4f*)d_ws;   // 99 * 16 B of scratch

    sde_setup<<<1, 128, 0, stream>>>(ts, cst);

    const int nthreads = BD / VEC;              // 262,144
    const int nblocks  = nthreads / BLOCK;      // 1024
    sde_main<<<nblocks, BLOCK, 0, stream>>>(y0, noise, theta, cst, out);
}